// HierGroupMultiLabelCE_12128987644155
// MI455X (gfx1250) — compile-verified
//
#include <hip/hip_runtime.h>
#include <hip/hip_bf16.h>
#include <stdint.h>

#define NC     19
#define NSP    2048
#define NSMALL 8192
#define NB     4
#define HW     (512*512)
#define TILE   256
#define EPSF   1e-8f

typedef uint32_t u32;
typedef uint64_t u64;
typedef unsigned long long ull;
typedef __attribute__((ext_vector_type(4))) u32   v4u;
typedef __attribute__((ext_vector_type(8))) int   v8i;
typedef __attribute__((ext_vector_type(4))) int   v4i;
typedef __attribute__((ext_vector_type(2))) float v2f;
typedef __attribute__((ext_vector_type(8))) float v8f;

// ---------------- workspace layout (bytes) ----------------
// packed   : NB*NSP*NC   * 8 = 1,245,184   (u64: key<<32 | pix)
// small_sum: NB*NSMALL*NC* 4 = 2,490,368   (f32)
// size_sum : NB*NSMALL   * 4 =   131,072   (i32)
// acc      : loss f32, nv u32
#define WS_PACKED_OFF   0
#define WS_SMALLSUM_OFF (NB*NSP*NC*8)
#define WS_SIZESUM_OFF  (WS_SMALLSUM_OFF + NB*NSMALL*NC*4)
#define WS_ACC_OFF      (WS_SIZESUM_OFF + NB*NSMALL*4)
#define WS_TOTAL_WORDS  ((WS_ACC_OFF + 8 + 3) / 4)

// Hardware fp32 atomic add (global_atomic_add_f32), no CAS expansion.
__device__ __forceinline__ void atomicAddF32HW(float* p, float v) {
    unsafeAtomicAdd(p, v);
}

template <typename T>
__device__ __forceinline__ T ntload(const T* p) {   // TH=NT streaming load
    return __builtin_nontemporal_load(p);
}

// ---------------- kernel A: zero workspace ----------------
__global__ __launch_bounds__(256) void k_init(u32* __restrict__ ws, int nwords) {
    for (int i = blockIdx.x * 256 + threadIdx.x; i < nwords; i += gridDim.x * 256)
        ws[i] = 0u;
}

// ---------------- kernel B: fused softmax + segmented argmax/sum scatter --
// one workgroup = 256 pixels of one sample; TDM stages the 19 class planes
// (stride HW) for those pixels into LDS.
__global__ __launch_bounds__(256) void k_scatter(
    const float*         __restrict__ logits,   // (N,C,H,W)
    const unsigned char* __restrict__ mask,     // (N,HW) bool
    const int*           __restrict__ big,      // (N,HW)
    const int*           __restrict__ small,    // (N,HW)
    const float*         __restrict__ noise,    // (N,HW,C)
    ull*                 __restrict__ packed,   // (N,NSP,NC)
    float*               __restrict__ small_sum,// (N,NSMALL,NC)
    int*                 __restrict__ size_sum) // (N,NSMALL)
{
    __shared__ float shl[NC * TILE];            // 19,456 B

    const int b        = blockIdx.x;
    const int n        = b / (HW / TILE);
    const int blockPix = (b % (HW / TILE)) * TILE;
    const int tid      = threadIdx.x;

#if __has_builtin(__builtin_amdgcn_tensor_load_to_lds)
    if (tid < 32) {  // wave 0 issues the tensor DMA (EXEC-independent op)
        const u64 gaddr = (u64)(uintptr_t)(logits + (size_t)n * NC * HW + blockPix);
        const u32 laddr = (u32)(uintptr_t)&shl[0];   // low 32 bits of flat = LDS byte addr
        v4u g0;
        g0[0] = 1u;                                   // count=1 valid, no gather
        g0[1] = laddr;                                // lds_addr
        g0[2] = (u32)gaddr;                           // global_addr[31:0]
        g0[3] = (u32)((gaddr >> 32) & 0x01FFFFFFu) | (2u << 30); // addr[56:32] | type=2
        const u32 td0 = (u32)HW, td1 = (u32)NC, t0 = (u32)TILE, t1 = (u32)NC;
        v8i g1;
        g1[0] = (int)(2u << 16);                      // wg_mask=0, data_size=4B
        g1[1] = (int)((td0 & 0xFFFFu) << 16);         // tensor_dim0[15:0]
        g1[2] = (int)((td0 >> 16) | (td1 << 16));     // tensor_dim0[31:16] | tensor_dim1[15:0]
        g1[3] = (int)((td1 >> 16) | (t0 << 16));      // tensor_dim1[31:16] | tile_dim0
        g1[4] = (int)t1;                              // tile_dim1 | tile_dim2=0
        g1[5] = (int)td0;                             // tensor_dim0_stride[31:0] = HW
        g1[6] = 0;                                    // stride0[47:32] | stride1[15:0]
        g1[7] = 0;                                    // stride1[47:16]
        const v4i gz4 = {0, 0, 0, 0};
        const v8i gz8 = {0, 0, 0, 0, 0, 0, 0, 0};
        __builtin_amdgcn_tensor_load_to_lds(g0, g1, gz4, gz4, gz8, 0);
        __builtin_amdgcn_s_wait_tensorcnt(0);
    }
    __syncthreads();
#else
    for (int c = tid; c < NC * TILE; c += 256)
        shl[c] = logits[(size_t)n * NC * HW + (size_t)(c / TILE) * HW + blockPix + (c % TILE)];
    __syncthreads();
#endif

    const int gpix = n * HW + blockPix + tid;
    if (ntload(mask + gpix)) {
        float l[NC];
        #pragma unroll
        for (int c = 0; c < NC; ++c) l[c] = shl[c * TILE + tid];

        float m = l[0];
        #pragma unroll
        for (int c = 1; c < NC; ++c) m = fmaxf(m, l[c]);
        float Z = 0.f;
        #pragma unroll
        for (int c = 0; c < NC; ++c) Z += __expf(l[c] - m);
        const float invZ = 1.f / Z;

        const int pix = blockPix + tid;
        const int bg  = ntload(big + gpix);
        const int sm  = ntload(small + gpix);
        const float* np = noise + (size_t)gpix * NC;
        ull*   pbase = packed    + ((size_t)n * NSP    + bg) * NC;
        float* sbase = small_sum + ((size_t)n * NSMALL + sm) * NC;

        #pragma unroll
        for (int c = 0; c < NC; ++c) {
            // segmented argmax: order-preserving float->u32, pack with pixel id.
            const float sc = l[c] + ntload(np + c);
            const u32 fb  = __float_as_uint(sc);
            const u32 key = (fb & 0x80000000u) ? ~fb : (fb | 0x80000000u);
            const ull pk  = ((ull)key << 32) | (u32)pix;
            atomicMax(pbase + c, pk);
            // segmented sum of -log(softmax + eps)
            const float nl = -__logf(__expf(l[c] - m) * invZ + EPSF);
            atomicAddF32HW(sbase + c, nl);
        }
        atomicAdd(size_sum + n * NSMALL + sm, 1);
    }
}

// ---------------- kernel C: gather + weighted contraction (WMMA reduce) --
__global__ __launch_bounds__(256) void k_finalize(
    const float* __restrict__ targets,   // (N,NSP,NC+1)
    const int*   __restrict__ small,     // (N,HW)
    const ull*   __restrict__ packed,
    const float* __restrict__ small_sum,
    const int*   __restrict__ size_sum,
    float*       __restrict__ lossAcc,
    unsigned*    __restrict__ nvAcc)
{
    const int idx = blockIdx.x * 256 + threadIdx.x;  // exactly NB*NSP threads
    const int n   = idx / NSP;

    const float* trg = targets + (size_t)idx * (NC + 1);
    float t[NC], ts = 0.f;
    #pragma unroll
    for (int c = 0; c < NC; ++c) { t[c] = ntload(trg + c); ts += t[c]; }

    const ull* pb = packed + (size_t)idx * NC;
    const bool keep = (pb[0] != 0ull) && (ts > 1.0f);  // nonempty & multi-label

    float    loss = 0.f;
    unsigned nv   = 0u;
    if (keep) {
        #pragma unroll
        for (int c = 0; c < NC; ++c) {
            if (t[c] != 0.f) {
                const int pix = (int)(u32)pb[c];
                const int sel = small[(size_t)n * HW + pix];
                loss += small_sum[((size_t)n * NSMALL + sel) * NC + c];
                nv   += (unsigned)size_sum[n * NSMALL + sel];
            }
        }
    }

    // wave32 reduction of `loss`; route the adds through the fp32 matrix unit:
    // A(16x4) holds one partial per lane, B = ones -> sum(D) = 16 * sum(A).
#if __has_builtin(__builtin_amdgcn_wmma_f32_16x16x4_f32)
    v2f a;  a[0] = loss; a[1] = 0.f;
    v2f bb; bb[0] = 1.f; bb[1] = 1.f;
    v8f d = {0.f, 0.f, 0.f, 0.f, 0.f, 0.f, 0.f, 0.f};
    d = __builtin_amdgcn_wmma_f32_16x16x4_f32(false, a, false, bb,
                                              (short)0, d, false, false);
    float s = d[0] + d[1] + d[2] + d[3] + d[4] + d[5] + d[6] + d[7];
    #pragma unroll
    for (int o = 16; o > 0; o >>= 1) s += __shfl_down(s, o, 32);
    const float waveLoss = s * 0.0625f;   // /16 (each B column repeats the row sums)
#else
    float s = loss;
    #pragma unroll
    for (int o = 16; o > 0; o >>= 1) s += __shfl_down(s, o, 32);
    const float waveLoss = s;
#endif
    unsigned nvs = nv;
    #pragma unroll
    for (int o = 16; o > 0; o >>= 1) nvs += __shfl_down(nvs, o, 32);

    if ((threadIdx.x & 31) == 0) {
        atomicAddF32HW(lossAcc, waveLoss);
        atomicAdd(nvAcc, nvs);
    }
}

// ---------------- kernel D: scalar divide ----------------
__global__ void k_finish(const float* __restrict__ lossAcc,
                         const unsigned* __restrict__ nvAcc,
                         float* __restrict__ out)
{
    if (threadIdx.x == 0 && blockIdx.x == 0)
        out[0] = lossAcc[0] / (1.0f + (float)nvAcc[0]);
}

extern "C" void kernel_launch(void* const* d_in, const int* in_sizes, int n_in,
                              void* d_out, int out_size, void* d_ws, size_t ws_size,
                              hipStream_t stream) {
    const float*         logits = (const float*)d_in[0];          // (N,C,H,W)
    const float*         trg    = (const float*)d_in[1];          // (N,NSP,NC+1)
    const unsigned char* mask   = (const unsigned char*)d_in[2];  // (N,H,W) bool
    const int*           big    = (const int*)d_in[3];            // (N,H,W)
    const int*           small  = (const int*)d_in[4];            // (N,H,W)
    const float*         noise  = (const float*)d_in[5];          // (N,HW,C)
    float*               out    = (float*)d_out;

    char* ws = (char*)d_ws;
    ull*      packed    = (ull*)(ws + WS_PACKED_OFF);
    float*    small_sum = (float*)(ws + WS_SMALLSUM_OFF);
    int*      size_sum  = (int*)(ws + WS_SIZESUM_OFF);
    float*    lossAcc   = (float*)(ws + WS_ACC_OFF);
    unsigned* nvAcc     = (unsigned*)(ws + WS_ACC_OFF + 4);

    k_init<<<1024, 256, 0, stream>>>((u32*)d_ws, (int)WS_TOTAL_WORDS);

    k_scatter<<<NB * (HW / TILE), 256, 0, stream>>>(
        logits, mask, big, small, noise, packed, small_sum, size_sum);

    k_finalize<<<(NB * NSP) / 256, 256, 0, stream>>>(
        trg, small, packed, small_sum, size_sum, lossAcc, nvAcc);

    k_finish<<<1, 1, 0, stream>>>(lossAcc, nvAcc, out);
}